// VectorQuantizer_1365799600183
// MI455X (gfx1250) — compile-verified
//
#include <hip/hip_runtime.h>
#include <hip/hip_bf16.h>

typedef __attribute__((ext_vector_type(2))) float v2f;
typedef __attribute__((ext_vector_type(8))) float v8f;

#define N_TOT    32768      // 32 * 32 * 32 rows
#define N_E      1024
#define E_DIM    64
#define HW       1024       // 32*32 spatial per batch
#define ZQ_OFF   1
#define PERP_OFF 2097153
#define ENC_OFF  2097154
#define IDX_OFF  35651586

// ---------------------------------------------------------------------------
// Kernel 1: codebook norms ||e||^2, zero histogram + loss accumulator
// ---------------------------------------------------------------------------
__global__ __launch_bounds__(1024)
void vq_prep(const float* __restrict__ emb, float* __restrict__ nrm,
             unsigned int* __restrict__ cnt, float* __restrict__ lsum)
{
    const int n = threadIdx.x;            // one thread per codebook entry
    const float* row = emb + n * E_DIM;
    float s = 0.f;
    #pragma unroll 8
    for (int c = 0; c < E_DIM; ++c) { float v = row[c]; s += v * v; }
    nrm[n] = s;
    cnt[n] = 0u;
    if (n == 0) *lsum = 0.f;
}

// ---------------------------------------------------------------------------
// Kernel 2: WMMA distance GEMM + argmin + all big outputs.
// One wave per 32-row super-tile (two 16-row WMMA tiles register-blocked,
// so each B fragment load feeds TWO wmma chains). 1024 waves total.
// Block = 128 thr = 4 waves.
// ---------------------------------------------------------------------------
__global__ __launch_bounds__(128)
void vq_main(const float* __restrict__ z, const float* __restrict__ emb,
             const float* __restrict__ nrm, unsigned int* __restrict__ cnt,
             float* __restrict__ lsum, float* __restrict__ out)
{
    __shared__ int idxs[4][32];

    const int lane = threadIdx.x & 31;
    const int wv   = threadIdx.x >> 5;
    const int pair = blockIdx.x * 4 + wv;          // 0..1023
    const int row0 = pair * 32;                    // 32 rows per wave
    const int r    = lane & 15;
    const int hi   = lane >> 4;
    const int zb   = (row0 >> 10) * (E_DIM * HW);  // batch offset in z (BCHW)
    const int hw0  = row0 & (HW - 1);              // spatial offset (32-aligned)

    // ---- Two A tiles: zf[row0..row0+15] and zf[row0+16..row0+31], WMMA
    // 16x4-f32 fragment layout: afragX[kc] = { zf[row][4kc+2hi], ...+1 }
    v2f afrag0[16], afrag1[16];
    {
        const float* zp = z + zb + hw0 + r;
        #pragma unroll
        for (int kc = 0; kc < 16; ++kc) {
            const int c0 = kc * 4 + 2 * hi;
            afrag0[kc].x = zp[(size_t)c0 * HW];
            afrag0[kc].y = zp[(size_t)(c0 + 1) * HW];
            afrag1[kc].x = zp[(size_t)c0 * HW + 16];
            afrag1[kc].y = zp[(size_t)(c0 + 1) * HW + 16];
        }
    }

    float minv0[8], minv1[8];
    int   mini0[8], mini1[8];
    #pragma unroll
    for (int v = 0; v < 8; ++v) {
        minv0[v] = 3.0e38f; mini0[v] = 0;
        minv1[v] = 3.0e38f; mini1[v] = 0;
    }

    // ---- sweep codebook in 16-wide tiles; each bfrag feeds 2 wmma ----
    for (int t = 0; t < 64; ++t) {
        const int n  = t * 16 + r;                  // this lane's column
        const float* bb = emb + (size_t)n * E_DIM + 2 * hi;
        v8f acc0 = {}, acc1 = {};
        #pragma unroll
        for (int kc = 0; kc < 16; ++kc) {
            v2f bfrag = *(const v2f*)(bb + kc * 4); // b64, 8B-aligned
            acc0 = __builtin_amdgcn_wmma_f32_16x16x4_f32(
                false, afrag0[kc], false, bfrag, (short)0, acc0, false, false);
            acc1 = __builtin_amdgcn_wmma_f32_16x16x4_f32(
                false, afrag1[kc], false, bfrag, (short)0, acc1, false, false);
        }
        const float en = nrm[n];
        #pragma unroll
        for (int v = 0; v < 8; ++v) {               // row = v + 8*hi (+16)
            const float s0 = en - 2.0f * acc0[v];   // + ||z||^2 is row-const
            if (s0 < minv0[v]) { minv0[v] = s0; mini0[v] = n; }
            const float s1 = en - 2.0f * acc1[v];
            if (s1 < minv1[v]) { minv1[v] = s1; mini1[v] = n; }
        }
    }

    // ---- cross-lane (16-lane half) min+index reduction ----
    #pragma unroll
    for (int v = 0; v < 8; ++v) {
        float bv0 = minv0[v]; int bi0 = mini0[v];
        float bv1 = minv1[v]; int bi1 = mini1[v];
        #pragma unroll
        for (int m = 1; m < 16; m <<= 1) {
            const float ov0 = __shfl_xor(bv0, m, 32);
            const int   oi0 = __shfl_xor(bi0, m, 32);
            if (ov0 < bv0 || (ov0 == bv0 && oi0 < bi0)) { bv0 = ov0; bi0 = oi0; }
            const float ov1 = __shfl_xor(bv1, m, 32);
            const int   oi1 = __shfl_xor(bi1, m, 32);
            if (ov1 < bv1 || (ov1 == bv1 && oi1 < bi1)) { bv1 = ov1; bi1 = oi1; }
        }
        mini0[v] = bi0; mini1[v] = bi1;
    }
    if (r == 0) {                       // lane 0 -> rows 0-7, lane 16 -> 8-15
        #pragma unroll
        for (int v = 0; v < 8; ++v) {
            idxs[wv][hi * 8 + v]      = mini0[v];
            idxs[wv][16 + hi * 8 + v] = mini1[v];
        }
    }
    __syncthreads();

    // ---- z_q gather (BCHW) + loss accumulation ----
    // lane (r,hi): rows row0+r and row0+16+r, channels c == hi (mod 2)
    {
        float ls = 0.f;
        #pragma unroll
        for (int h2 = 0; h2 < 2; ++h2) {
            const int id = idxs[wv][h2 * 16 + r];
            const float* erow = emb + (size_t)id * E_DIM;
            const int hw = hw0 + h2 * 16 + r;
            #pragma unroll 8
            for (int c0 = 0; c0 < E_DIM; c0 += 2) {
                const int c = c0 + hi;
                const float e  = erow[c];
                const size_t o = (size_t)zb + (size_t)c * HW + hw;
                const float zv = z[o];
                const float d  = e - zv;
                ls += d * d;
                out[ZQ_OFF + o] = e;
            }
        }
        #pragma unroll
        for (int m = 1; m < 32; m <<= 1) ls += __shfl_xor(ls, m, 32);
        if (lane == 0) atomicAdd(lsum, ls);
    }

    // ---- histogram + indices (32 rows -> all 32 lanes) ----
    {
        const int id = idxs[wv][lane];
        atomicAdd(&cnt[id], 1u);
        out[IDX_OFF + row0 + lane] = (float)id;
    }

    // ---- one-hot encodings: 32 rows x 1024 floats, b64 stores ----
    float* encBase = out + ENC_OFF;
    for (int rr = 0; rr < 32; ++rr) {
        const int id = idxs[wv][rr];
        float* rowp = encBase + (size_t)(row0 + rr) * N_E;
        #pragma unroll
        for (int j = 0; j < 16; ++j) {
            const int col = j * 64 + lane * 2;
            float2 vv;
            vv.x = (col     == id) ? 1.0f : 0.0f;
            vv.y = (col + 1 == id) ? 1.0f : 0.0f;
            *(float2*)(rowp + col) = vv;            // 8B aligned
        }
    }
}

// ---------------------------------------------------------------------------
// Kernel 3: perplexity + loss finalize
// ---------------------------------------------------------------------------
__global__ __launch_bounds__(1024)
void vq_fin(const unsigned int* __restrict__ cnt,
            const float* __restrict__ lsum, float* __restrict__ out)
{
    __shared__ float red[1024];
    const int k = threadIdx.x;
    const float p = (float)cnt[k] * (1.0f / (float)N_TOT);
    red[k] = p * logf(p + 1e-10f);
    __syncthreads();
    for (int s = 512; s > 0; s >>= 1) {
        if (k < s) red[k] += red[k + s];
        __syncthreads();
    }
    if (k == 0) {
        out[PERP_OFF] = expf(-red[0]);
        // loss = (1 + beta) * mean((z_q - z)^2), beta = 0.25
        out[0] = 1.25f * (*lsum) * (1.0f / (float)(N_TOT * E_DIM));
    }
}

// ---------------------------------------------------------------------------
extern "C" void kernel_launch(void* const* d_in, const int* in_sizes, int n_in,
                              void* d_out, int out_size, void* d_ws, size_t ws_size,
                              hipStream_t stream) {
    const float* z   = (const float*)d_in[0];
    const float* emb = (const float*)d_in[1];
    float* out = (float*)d_out;

    float*        nrm = (float*)d_ws;
    unsigned int* cnt = (unsigned int*)(nrm + N_E);
    float*        ls  = (float*)(cnt + N_E);

    vq_prep<<<1, 1024, 0, stream>>>(emb, nrm, cnt, ls);
    vq_main<<<256, 128, 0, stream>>>(z, emb, nrm, cnt, ls, out);
    vq_fin<<<1, 1024, 0, stream>>>(cnt, ls, out);
}